// Net_28424093565711
// MI455X (gfx1250) — compile-verified
//
#include <hip/hip_runtime.h>
#include <math.h>

#define NL    5
#define HID   3
#define NB    10
#define TSEQ  131072
#define CHUNK 64
#define NCHUNK (TSEQ / CHUNK)   // 2048 exactly
#define RING  128               // 2 chunks of 64 steps, 16B/step

typedef __attribute__((ext_vector_type(2))) float v2f;
typedef __attribute__((ext_vector_type(3))) float v3f;
typedef __attribute__((ext_vector_type(4))) float v4f;
typedef __attribute__((ext_vector_type(8))) float v8f;

// ---------------------------------------------------------------------------
// Cross-lane gathers. Both required permutations are static and identical in
// each 16-lane row, so V_PERMLANE16_B32 (pure VALU, no LDS crossbar latency)
// is the ideal tool. Nibble L of {s2,s1} selects the source lane for lane L.
//   own  triple: lane L reads lane 3*(L/3)+k
//   prev triple: lane L reads lane max(3*(L/3)-3,0)+k   (layer0 result unused)
// ---------------------------------------------------------------------------
#if __has_builtin(__builtin_amdgcn_permlane16)
#define PERM16(v, s1, s2) __builtin_amdgcn_permlane16((v), (v), (s1), (s2), 0, 0)
#elif __has_builtin(__builtin_amdgcn_permlane16_b32)
#define PERM16(v, s1, s2) __builtin_amdgcn_permlane16_b32((v), (v), (s1), (s2), 0, 0)
#endif

__device__ __forceinline__ float lane_gather(float v, int srcLane) {
    return __int_as_float(__builtin_amdgcn_ds_bpermute(srcLane << 2, __float_as_int(v)));
}
#ifdef PERM16
__device__ __forceinline__ float perm16f(float v, unsigned s1, unsigned s2) {
    return __int_as_float((int)PERM16((unsigned)__float_as_int(v), s1, s2));
}
#endif

__device__ __forceinline__ float sigmoid_fast(float x) {
    float e = __expf(-x);                       // v_exp_f32 (TRANS)
    return __builtin_amdgcn_rcpf(1.0f + e);     // v_rcp_f32
}
__device__ __forceinline__ float tanh_fast(float x) {
    float e = __expf(-2.0f * x);
    return (1.0f - e) * __builtin_amdgcn_rcpf(1.0f + e);
}

// ---------------------------------------------------------------------------
// Kernel 1: 5-layer wavefront-pipelined LSTM scan.
// grid = 5 blocks x 1 wave32. Each half-wave owns one batch element; lane
// (half*16 + layer*3 + j) owns hidden unit j of pipeline stage `layer`.
// Layer l processes timestep t = n - l at pipeline iteration n; h flows
// between stages via in-wave lane permutes (no barriers, no state in LDS).
// x is streamed through a 128-step LDS ring (16B/step), loaded two chunks
// ahead with global_load_b128 so HBM latency never touches the chain.
// ---------------------------------------------------------------------------
__global__ __launch_bounds__(32) void lstm_pipeline_kernel(
    const float* __restrict__ x_input, const float* __restrict__ h0,
    const float* __restrict__ c0,      const float* __restrict__ W_ih,
    const float* __restrict__ W_hh,    const float* __restrict__ b_ih,
    const float* __restrict__ b_hh,    float* __restrict__ h_final)
{
    __shared__ __align__(16) float ring[2][RING * 4];   // [half][slot*4 + k]

    const int lane  = threadIdx.x;
    const int half  = lane >> 4;
    const int li    = lane & 15;
    const int layer = (li < 15) ? (li / 3) : (NL - 1);  // lanes 15/31 idle: clamp
    const int j     = (li < 15) ? (li % 3) : 0;
    const int bat   = blockIdx.x * 2 + half;

    // ---- register-resident weights: rows {j, 3+j, 6+j, 9+j} of this layer ----
    float wi[4][3], wh[4][3], bias[4];
#pragma unroll
    for (int g = 0; g < 4; ++g) {
        const int row = g * HID + j;
#pragma unroll
        for (int k = 0; k < HID; ++k) {
            wi[g][k] = W_ih[(layer * 4 * HID + row) * HID + k];
            wh[g][k] = W_hh[(layer * 4 * HID + row) * HID + k];
        }
        bias[g] = b_ih[layer * 4 * HID + row] + b_hh[layer * 4 * HID + row];
    }

    float h = h0[(layer * NB + bat) * HID + j];
    float c = c0[(layer * NB + bat) * HID + j];

    const int base_own  = (lane & 16) + layer * 3;       // bpermute fallback only
    const int base_prev = (layer > 0) ? (base_own - 3) : base_own;
    (void)base_own; (void)base_prev;

    float ho0, ho1, ho2, hp0, hp1, hp2;
#ifdef PERM16
#define GATHER_ALL() do {                                  \
        ho0 = perm16f(h, 0x66333000u, 0xFCCC9996u);        \
        ho1 = perm16f(h, 0x77444111u, 0xFDDDAAA7u);        \
        ho2 = perm16f(h, 0x88555222u, 0xFEEEBBB8u);        \
        hp0 = perm16f(h, 0x33000000u, 0xF9996663u);        \
        hp1 = perm16f(h, 0x44111111u, 0xFAAA7774u);        \
        hp2 = perm16f(h, 0x55222222u, 0xFBBB8885u);        \
    } while (0)
#else
#define GATHER_ALL() do {                                  \
        ho0 = lane_gather(h, base_own + 0);                \
        ho1 = lane_gather(h, base_own + 1);                \
        ho2 = lane_gather(h, base_own + 2);                \
        hp0 = lane_gather(h, base_prev + 0);               \
        hp1 = lane_gather(h, base_prev + 1);               \
        hp2 = lane_gather(h, base_prev + 2);               \
    } while (0)
#endif
    GATHER_ALL();

    // ---- x streaming machinery ----
    const float* xg = x_input + (size_t)bat * TSEQ * HID;
    v4f r0, r1, r2;   // in-flight chunk: lane li holds floats [li*12, li*12+12)
    auto load_chunk = [&](int q) {
        const v4f* p = (const v4f*)(xg + (size_t)q * CHUNK * HID) + li * 3;
        r0 = p[0]; r1 = p[1]; r2 = p[2];
    };
    // repack 12 packed floats -> 4 ring slots of 16B (steps 4*li .. 4*li+3)
    auto store_chunk = [&](int base_slot) {
        float* rb = &ring[half][(base_slot + 4 * li) * 4];
        *(v3f*)(rb +  0) = (v3f){r0.x, r0.y, r0.z};
        *(v3f*)(rb +  4) = (v3f){r0.w, r1.x, r1.y};
        *(v3f*)(rb +  8) = (v3f){r1.z, r1.w, r2.x};
        *(v3f*)(rb + 12) = (v3f){r2.y, r2.z, r2.w};
    };

    auto lstm_step = [&](float in0, float in1, float in2, bool valid) {
        float gi = bias[0] + wi[0][0]*in0 + wi[0][1]*in1 + wi[0][2]*in2
                           + wh[0][0]*ho0 + wh[0][1]*ho1 + wh[0][2]*ho2;
        float gf = bias[1] + wi[1][0]*in0 + wi[1][1]*in1 + wi[1][2]*in2
                           + wh[1][0]*ho0 + wh[1][1]*ho1 + wh[1][2]*ho2;
        float gg = bias[2] + wi[2][0]*in0 + wi[2][1]*in1 + wi[2][2]*in2
                           + wh[2][0]*ho0 + wh[2][1]*ho1 + wh[2][2]*ho2;
        float go = bias[3] + wi[3][0]*in0 + wi[3][1]*in1 + wi[3][2]*in2
                           + wh[3][0]*ho0 + wh[3][1]*ho1 + wh[3][2]*ho2;
        float ia = sigmoid_fast(gi);
        float fa = sigmoid_fast(gf);
        float ga = tanh_fast(gg);
        float oa = sigmoid_fast(go);
        float cn = fa * c + ia * ga;
        float hn = oa * tanh_fast(cn);
        c = valid ? cn : c;
        h = valid ? hn : h;
        GATHER_ALL();
    };

    // ---- prologue: ring[0..63] <- chunk0 (one-time stall), regs <- chunk1 ----
    load_chunk(0);
    store_chunk(0);
    load_chunk(1);

    float xc0 = 0.f, xc1 = 0.f, xc2 = 0.f, xn0, xn1, xn2;
    {
        v3f xv = *(const v3f*)&ring[half][0];
        xn0 = xv.x; xn1 = xv.y; xn2 = xv.z;              // x[0] prefetched
    }

    // ---- chunk 0: pipeline fill (mask iterations n < layer) ----
    {
        store_chunk(CHUNK);                              // chunk 1 -> slots 64..127
        load_chunk(2);
        for (int i = 0; i < CHUNK; ++i) {
            xc0 = xn0; xc1 = xn1; xc2 = xn2;
            v3f xv = *(const v3f*)&ring[half][((i + 1) & (RING - 1)) * 4];
            xn0 = xv.x; xn1 = xv.y; xn2 = xv.z;
            float in0 = (layer == 0) ? xc0 : hp0;
            float in1 = (layer == 0) ? xc1 : hp1;
            float in2 = (layer == 0) ? xc2 : hp2;
            lstm_step(in0, in1, in2, i >= layer);
        }
    }

    // ---- steady state: all layers valid, mask folds away ----
    for (int q = 1; q < NCHUNK; ++q) {
        store_chunk(((q + 1) & 1) * CHUNK);              // chunk q+1 -> other half
        const int q2 = (q + 2 < NCHUNK) ? (q + 2) : (NCHUNK - 1);
        load_chunk(q2);                                  // prefetch chunk q+2
        const int nbase = q * CHUNK;
        for (int i = 0; i < CHUNK; ++i) {
            xc0 = xn0; xc1 = xn1; xc2 = xn2;
            v3f xv = *(const v3f*)&ring[half][((nbase + i + 1) & (RING - 1)) * 4];
            xn0 = xv.x; xn1 = xv.y; xn2 = xv.z;
            float in0 = (layer == 0) ? xc0 : hp0;
            float in1 = (layer == 0) ? xc1 : hp1;
            float in2 = (layer == 0) ? xc2 : hp2;
            lstm_step(in0, in1, in2, true);
        }
    }

    // ---- drain: 4 iterations; layer l valid iff d < l (x path is dead) ----
    for (int d = 0; d < NL - 1; ++d) {
        float in0 = (layer == 0) ? xc0 : hp0;
        float in1 = (layer == 0) ? xc1 : hp1;
        float in2 = (layer == 0) ? xc2 : hp2;
        lstm_step(in0, in1, in2, d < layer);
    }

    if (li >= 12 && li < 15) {                           // layer-4 lanes: h[T-1]
        h_final[bat * HID + (li - 12)] = h;
    }
}

// ---------------------------------------------------------------------------
// Kernel 2: head.  z = h_final @ W_lin^T + b_lin via V_WMMA_F32_16X16X4_F32,
// then relu + softmax(axis=1).  Reads h_final from `out`, overwrites `out`.
// Branch-free value construction; EXEC is all ones at the WMMA.
// ---------------------------------------------------------------------------
__global__ __launch_bounds__(32) void head_kernel(
    const float* __restrict__ W_lin, const float* __restrict__ b_lin,
    float* __restrict__ out)
{
    __shared__ float z[16][16];
    const int  lane = threadIdx.x;
    const int  m    = lane & 15;
    const bool hi   = lane >= 16;

    // A (16x4 f32): lanes 0-15 hold K=0,1 ; lanes 16-31 hold K=2,3 ; M = lane&15
    const int  mc  = (m < NB) ? m : 0;
    const bool mok = (m < NB);
    const float hv0 = out[mc * HID + 0];
    const float hv1 = out[mc * HID + 1];
    const float hv2 = out[mc * HID + 2];
    v2f A;
    A.x = mok ? (hi ? hv2 : hv0) : 0.0f;
    A.y = (mok && !hi) ? hv1 : 0.0f;          // K=3 column is zero padding

    // B (4x16): B[k][n] = W_lin[n*HID+k]; same K split across half-waves, N = lane&15
    const int  nc  = (m < HID) ? m : 0;
    const bool nok = (m < HID);
    const float w0 = W_lin[nc * HID + 0];
    const float w1 = W_lin[nc * HID + 1];
    const float w2 = W_lin[nc * HID + 2];
    v2f B;
    B.x = nok ? (hi ? w2 : w0) : 0.0f;
    B.y = (nok && !hi) ? w1 : 0.0f;

    // C (16x16 f32): VGPR r -> M = r (+8 for hi lanes), N = lane&15 ; C = bias
    const float bl = nok ? b_lin[nc] : 0.0f;
    v8f C;
#pragma unroll
    for (int r = 0; r < 8; ++r) {
        const int M = r + (hi ? 8 : 0);
        C[r] = (nok && M < NB) ? bl : 0.0f;
    }

    v8f D = __builtin_amdgcn_wmma_f32_16x16x4_f32(
        /*neg_a=*/false, A, /*neg_b=*/false, B,
        /*c_mod=*/(short)0, C, /*reuse_a=*/false, /*reuse_b=*/false);

#pragma unroll
    for (int r = 0; r < 8; ++r) {
        const int M = r + (hi ? 8 : 0);
        z[M][m] = D[r];
    }
    __syncthreads();

    if (lane < NB * HID) {
        const int b = lane / HID, k = lane % HID;
        float z0 = fmaxf(z[b][0], 0.0f);
        float z1 = fmaxf(z[b][1], 0.0f);
        float z2 = fmaxf(z[b][2], 0.0f);
        float mx = fmaxf(z0, fmaxf(z1, z2));
        float e0 = __expf(z0 - mx), e1 = __expf(z1 - mx), e2 = __expf(z2 - mx);
        float s  = e0 + e1 + e2;
        float ek = (k == 0) ? e0 : ((k == 1) ? e1 : e2);
        out[lane] = ek / s;
    }
}

extern "C" void kernel_launch(void* const* d_in, const int* in_sizes, int n_in,
                              void* d_out, int out_size, void* d_ws, size_t ws_size,
                              hipStream_t stream) {
    (void)in_sizes; (void)n_in; (void)d_ws; (void)ws_size; (void)out_size;
    const float* x_input = (const float*)d_in[0];
    const float* h0      = (const float*)d_in[1];
    const float* c0      = (const float*)d_in[2];
    const float* W_ih    = (const float*)d_in[3];
    const float* W_hh    = (const float*)d_in[4];
    const float* b_ih    = (const float*)d_in[5];
    const float* b_hh    = (const float*)d_in[6];
    const float* W_lin   = (const float*)d_in[7];
    const float* b_lin   = (const float*)d_in[8];
    float* out = (float*)d_out;

    lstm_pipeline_kernel<<<NB / 2, 32, 0, stream>>>(
        x_input, h0, c0, W_ih, W_hh, b_ih, b_hh, out);
    head_kernel<<<1, 32, 0, stream>>>(W_lin, b_lin, out);
}